// Transformer_46909632807248
// MI455X (gfx1250) — compile-verified
//
#include <hip/hip_runtime.h>
#include <math.h>

// ---------------------------------------------------------------- constants
#define E      512
#define LTOK   4096
#define NB     4
#define MS     12          // real slots
#define MPAD   16          // padded slot rows (full WMMA tiles)
#define DFF    2048
#define SCALING 0.04419417382415922f   // 512^-0.5
#define LN_EPS 1e-5f

typedef float v2f __attribute__((ext_vector_type(2)));
typedef float v8f __attribute__((ext_vector_type(8)));

// CDNA5 fp32 matrix core: D(16x16) = A(16x4) * B(4x16) + C
__device__ __forceinline__ v8f wmma4(v2f a, v2f b, v8f c) {
  return __builtin_amdgcn_wmma_f32_16x16x4_f32(
      /*neg_a=*/false, a, /*neg_b=*/false, b,
      /*c_mod=*/(short)0, c, /*reuse_a=*/false, /*reuse_b=*/false);
}

// CDNA5 async global->LDS copy, 16 bytes per lane (ASYNCcnt-tracked)
__device__ __forceinline__ void async_g2l_b128(unsigned lds_byte_off, const float* gaddr) {
  asm volatile("global_load_async_to_lds_b128 %0, %1, off"
               :: "v"(lds_byte_off), "v"(gaddr) : "memory");
}
__device__ __forceinline__ void wait_async0() {
  asm volatile("s_wait_asynccnt 0x0" ::: "memory");
}

// ---------------------------------------------------------------- GEMM, B = W.T
// D[r,c] = act( scale * sum_k A[r,k]*W[c,k] + bias[c] ) + addend[r,c]
// W is [O,K] row-major. One wave per 16x64 output block (4 WMMA tiles),
// pointer-bumped b64 loads, 4 accumulators for A reuse.
__global__ __launch_bounds__(32) void wmma_gemm_wt(
    const float* __restrict__ A, const float* __restrict__ B, float* __restrict__ D,
    const float* __restrict__ bias, const float* __restrict__ addend,
    int K, int lda, int ldb, int ldd, float scale, int act)
{
  const int n0 = blockIdx.x * 64;
  const int m0 = blockIdx.y * 16;
  const int lane = threadIdx.x;
  const int half = lane >> 4;       // K sub-offset: 0 -> k+0..1, 1 -> k+2..3
  const int lid  = lane & 15;

  const float* ar = A + (size_t)(m0 + lid) * lda + 2 * half;
  const float* b0 = B + (size_t)(n0 + lid) * ldb + 2 * half;
  const float* b1 = b0 + (size_t)16 * ldb;
  const float* b2 = b0 + (size_t)32 * ldb;
  const float* b3 = b0 + (size_t)48 * ldb;

  v8f c0 = {0.f,0.f,0.f,0.f,0.f,0.f,0.f,0.f};
  v8f c1 = c0, c2 = c0, c3 = c0;

#pragma unroll 2
  for (int kb = 0; kb < K; kb += 4) {
    const v2f a  = *(const v2f*)ar;
    const v2f q0 = *(const v2f*)b0;
    const v2f q1 = *(const v2f*)b1;
    const v2f q2 = *(const v2f*)b2;
    const v2f q3 = *(const v2f*)b3;
    c0 = wmma4(a, q0, c0);
    c1 = wmma4(a, q1, c1);
    c2 = wmma4(a, q2, c2);
    c3 = wmma4(a, q3, c3);
    ar += 4; b0 += 4; b1 += 4; b2 += 4; b3 += 4;
  }

  v8f acc[4] = {c0, c1, c2, c3};
#pragma unroll
  for (int t = 0; t < 4; ++t) {
    const int col = n0 + t * 16 + lid;
#pragma unroll
    for (int v = 0; v < 8; ++v) {
      const int r = m0 + v + 8 * half;          // C/D layout: row = v + 8*(lane>=16)
      float x = acc[t][v] * scale;
      if (bias)   x += bias[col];
      if (act==1) x = 0.5f * x * (1.0f + erff(x * 0.70710678118654752f)); // exact GELU
      if (addend) x += addend[(size_t)r * ldd + col];
      D[(size_t)r * ldd + col] = x;
    }
  }
}

// ---------------------------------------------------------------- GEMM, B K-major, batched
// D = A @ B + addend ; B(k,c) = B[k*ldb + c]. One wave per 16x16 tile.
__global__ __launch_bounds__(32) void wmma_gemm_nn_b(
    const float* __restrict__ A, const float* __restrict__ B, float* __restrict__ D,
    const float* __restrict__ addend,
    int K, int lda, int ldb, int ldd,
    long sAb, long sBb, long sDb, long sAddb)
{
  const int n0 = blockIdx.x * 16;
  const int m0 = blockIdx.y * 16;
  const int bb = blockIdx.z;
  const int lane = threadIdx.x;
  const int half = lane >> 4;
  const int lid  = lane & 15;

  const float* ar = A + (size_t)bb * sAb + (size_t)(m0 + lid) * lda + 2 * half;
  const float* bp = B + (size_t)bb * sBb + (size_t)(2 * half) * ldb + n0 + lid;
  const size_t bstep = (size_t)4 * ldb;

  v8f c = {0.f,0.f,0.f,0.f,0.f,0.f,0.f,0.f};
#pragma unroll 2
  for (int kb = 0; kb < K; kb += 4) {
    const v2f a = *(const v2f*)ar;
    v2f bv;  bv.x = bp[0];  bv.y = bp[ldb];
    c = wmma4(a, bv, c);
    ar += 4;  bp += bstep;
  }

  const int col = n0 + lid;
#pragma unroll
  for (int v = 0; v < 8; ++v) {
    const int r = m0 + v + 8 * half;
    float x = c[v];
    if (addend) x += addend[(size_t)bb * sAddb + (size_t)r * ldd + col];
    D[(size_t)bb * sDb + (size_t)r * ldd + col] = x;
  }
}

// ---------------------------------------------------------------- small helpers
__global__ void init_slots(const float* __restrict__ sv, float* __restrict__ slotsP) {
  int idx = blockIdx.x * blockDim.x + threadIdx.x;      // over NB*MPAD*E
  if (idx >= NB * MPAD * E) return;
  int e = idx % E, m = (idx / E) % MPAD, n = idx / (E * MPAD);
  slotsP[idx] = (m < MS) ? sv[((size_t)m * NB + n) * E + e] : 0.0f;   // [12,N,E] -> [N,16,E]
}

// att[n,m,l] = dot(Qs[n,m,:], Ks[n,l,:])   (Ks in [L,N,E] layout)
__global__ void slot_scores(const float* __restrict__ qs, const float* __restrict__ ks,
                            float* __restrict__ att) {
  __shared__ float sQ[MS][E];                           // 24 KB
  const int n = blockIdx.y;
  for (int i = threadIdx.x; i < MS * E; i += 256)
    sQ[i / E][i % E] = qs[((size_t)n * MPAD + i / E) * E + (i % E)];
  __syncthreads();
  const int l = blockIdx.x * 256 + threadIdx.x;
  float acc[MS];
#pragma unroll
  for (int m = 0; m < MS; ++m) acc[m] = 0.f;
  const float* kr = ks + ((size_t)l * NB + n) * E;
  for (int e = 0; e < E; ++e) {
    const float kv = kr[e];
#pragma unroll
    for (int m = 0; m < MS; ++m) acc[m] += sQ[m][e] * kv;
  }
#pragma unroll
  for (int m = 0; m < MS; ++m) att[((size_t)n * MPAD + m) * LTOK + l] = acc[m];
}

// softmax over l (4096) for one (n,m) row
__global__ void softmax_rows(const float* __restrict__ att, float* __restrict__ out) {
  __shared__ float red[256];
  const int n = blockIdx.x / MS, m = blockIdx.x % MS, tid = threadIdx.x;
  const float* row  = att + ((size_t)n * MPAD + m) * LTOK;
  float*       orow = out + ((size_t)n * MPAD + m) * LTOK;
  float mx = -INFINITY;
  for (int i = tid; i < LTOK; i += 256) mx = fmaxf(mx, row[i]);
  red[tid] = mx; __syncthreads();
  for (int s = 128; s > 0; s >>= 1) { if (tid < s) red[tid] = fmaxf(red[tid], red[tid+s]); __syncthreads(); }
  mx = red[0]; __syncthreads();
  float sum = 0.f;
  for (int i = tid; i < LTOK; i += 256) sum += expf(row[i] - mx);
  red[tid] = sum; __syncthreads();
  for (int s = 128; s > 0; s >>= 1) { if (tid < s) red[tid] += red[tid+s]; __syncthreads(); }
  const float inv = 1.0f / red[0];
  for (int i = tid; i < LTOK; i += 256) orow[i] = expf(row[i] - mx) * inv;
}

// slot_assign[n,m,l] = softmax over m(<12) of att[n,m,l]  -> d_out section [N,12,L]
__global__ void slot_assign_k(const float* __restrict__ att, float* __restrict__ sa) {
  const int n = blockIdx.y;
  const int l = blockIdx.x * 256 + threadIdx.x;
  float v[MS], mx = -INFINITY;
#pragma unroll
  for (int m = 0; m < MS; ++m) { v[m] = att[((size_t)n * MPAD + m) * LTOK + l]; mx = fmaxf(mx, v[m]); }
  float s = 0.f;
#pragma unroll
  for (int m = 0; m < MS; ++m) { v[m] = expf(v[m] - mx); s += v[m]; }
  const float inv = 1.0f / s;
#pragma unroll
  for (int m = 0; m < MS; ++m) sa[((size_t)n * MS + m) * LTOK + l] = v[m] * inv;
}

// layer norm over E=512, one block (256 thr) per row
__global__ void layernorm_k(const float* __restrict__ x, const float* __restrict__ g,
                            const float* __restrict__ b, float* __restrict__ y) {
  __shared__ float red[256];
  const int r = blockIdx.x, tid = threadIdx.x;
  const float* xr = x + (size_t)r * E;
  const float a0 = xr[tid], a1 = xr[tid + 256];
  red[tid] = a0 + a1; __syncthreads();
  for (int s = 128; s > 0; s >>= 1) { if (tid < s) red[tid] += red[tid+s]; __syncthreads(); }
  const float mean = red[0] * (1.0f / E); __syncthreads();
  const float d0 = a0 - mean, d1 = a1 - mean;
  red[tid] = d0 * d0 + d1 * d1; __syncthreads();
  for (int s = 128; s > 0; s >>= 1) { if (tid < s) red[tid] += red[tid+s]; __syncthreads(); }
  const float inv = rsqrtf(red[0] * (1.0f / E) + LN_EPS);
  y[(size_t)r * E + tid]       = d0 * inv * g[tid]       + b[tid];
  y[(size_t)r * E + tid + 256] = d1 * inv * g[tid + 256] + b[tid + 256];
}

// slot_att[n,l,m] = softmax_m dot(Qf[l,n,:], slots[n,m,:]) -> d_out section [N,L,12]
__global__ void slot_att_k(const float* __restrict__ Qf, const float* __restrict__ slotsP,
                           float* __restrict__ sat) {
  __shared__ float sS[MS][E];
  const int n = blockIdx.y;
  for (int i = threadIdx.x; i < MS * E; i += 256)
    sS[i / E][i % E] = slotsP[((size_t)n * MPAD + i / E) * E + (i % E)];
  __syncthreads();
  const int l = blockIdx.x * 256 + threadIdx.x;
  float acc[MS];
#pragma unroll
  for (int m = 0; m < MS; ++m) acc[m] = 0.f;
  const float* qr = Qf + ((size_t)l * NB + n) * E;
  for (int e = 0; e < E; ++e) {
    const float qv = qr[e];
#pragma unroll
    for (int m = 0; m < MS; ++m) acc[m] += qv * sS[m][e];
  }
  float mx = -INFINITY;
#pragma unroll
  for (int m = 0; m < MS; ++m) mx = fmaxf(mx, acc[m]);
  float s = 0.f;
#pragma unroll
  for (int m = 0; m < MS; ++m) { acc[m] = expf(acc[m] - mx); s += acc[m]; }
  const float inv = 1.0f / s;
#pragma unroll
  for (int m = 0; m < MS; ++m) sat[((size_t)n * LTOK + l) * MS + m] = acc[m] * inv;
}

__global__ void slots_out_k(const float* __restrict__ slotsP, float* __restrict__ o) {
  int idx = blockIdx.x * blockDim.x + threadIdx.x;      // over MS*NB*E, layout [12,N,E]
  if (idx >= MS * NB * E) return;
  int e = idx % E, n = (idx / E) % NB, m = idx / (E * NB);
  o[idx] = slotsP[((size_t)n * MPAD + m) * E + e];
}

// ---------------------------------------------------------------- flash attention
// One WG (8 waves) per (n, 16-row l-block). Full 16x4096 prob panel lives in LDS
// (256 KB of the 320 KB/WGP). logits = (Q K^T) * (slot_att @ slot_assign) via WMMA
// + rank-12 VALU modulation; softmax in LDS; O = P V via WMMA; divide by row sum.
// Q panel is staged with CDNA5 async global->LDS copies (ASYNCcnt).
__global__ __launch_bounds__(256) void flash_attn(
    const float* __restrict__ Qf,  // [L,N,E]
    const float* __restrict__ Kf,  // [L,N,E]
    const float* __restrict__ Vf,  // [L,N,E]
    const float* __restrict__ sat, // [N,L,12]
    const float* __restrict__ sas, // [N,12,L]
    float* __restrict__ out)       // [L,N,E]
{
  extern __shared__ float sm[];
  float* P   = sm;                       // 16*4096
  float* Qs  = P   + MPAD * LTOK;        // 16*512
  float* sA  = Qs  + MPAD * E;           // 16*12 slot_att rows
  float* wmx = sA  + MPAD * MS;          // 8*16 per-wave row max
  float* wsm = wmx + 8 * 16;             // 8*16 per-wave row sum

  const int n   = blockIdx.y;
  const int l0  = blockIdx.x * 16;
  const int tid = threadIdx.x;
  const int w    = tid >> 5;
  const int lane = tid & 31;
  const int half = lane >> 4;
  const int lid  = lane & 15;

  // --- stage Q panel (16x512 f32, 32KB) via async global->LDS, 16B per op.
  // Dynamic LDS begins right after static LDS: byte offset = groupstaticsize.
  {
    const unsigned qs_lds = __builtin_amdgcn_groupstaticsize()
                          + (unsigned)(MPAD * LTOK * sizeof(float));
    for (int i = tid; i < 16 * (E / 4); i += 256) {     // 2048 x 16B
      const int r  = i >> 7;                            // /(E/4)
      const int e4 = i & (E / 4 - 1);
      const float* g = Qf + ((size_t)(l0 + r) * NB + n) * E + e4 * 4;
      async_g2l_b128(qs_lds + (unsigned)((r * E + e4 * 4) * sizeof(float)), g);
    }
  }
  for (int i = tid; i < 16 * MS; i += 256) {
    const int r = i / MS, m = i - r * MS;
    sA[i] = sat[((size_t)n * LTOK + (l0 + r)) * MS + m];
  }
  wait_async0();
  __syncthreads();

  // ---- phase 1: logits for this wave's 512-column strip
  float rmax[8];
#pragma unroll
  for (int v = 0; v < 8; ++v) rmax[v] = -INFINITY;

  for (int st = 0; st < 32; ++st) {
    const int s0 = w * 512 + st * 16;
    v8f c = {0.f,0.f,0.f,0.f,0.f,0.f,0.f,0.f};
    const float* qp = Qs + (size_t)lid * E + 2 * half;                 // LDS
    const float* kp = Kf + ((size_t)(s0 + lid) * NB + n) * E + 2 * half;
#pragma unroll 4
    for (int kb = 0; kb < E; kb += 4) {
      const v2f a = *(const v2f*)qp;
      const v2f b = *(const v2f*)kp;
      c = wmma4(a, b, c);
      qp += 4;  kp += 4;
    }
    float scol[MS];
#pragma unroll
    for (int m = 0; m < MS; ++m) scol[m] = sas[((size_t)n * MS + m) * LTOK + s0 + lid];
#pragma unroll
    for (int v = 0; v < 8; ++v) {
      const int r = v + 8 * half;
      float ta = 0.f;
#pragma unroll
      for (int m = 0; m < MS; ++m) ta += sA[r * MS + m] * scol[m];
      const float val = c[v] * ta;                 // token_att * token_assign
      P[r * LTOK + s0 + lid] = val;
      rmax[v] = fmaxf(rmax[v], val);
    }
  }
#pragma unroll
  for (int v = 0; v < 8; ++v) {
    float x = rmax[v];
    for (int off = 1; off < 16; off <<= 1) x = fmaxf(x, __shfl_xor(x, off, 32));
    rmax[v] = x;
  }
  if (lid == 0) {
#pragma unroll
    for (int v = 0; v < 8; ++v) wmx[w * 16 + v + 8 * half] = rmax[v];
  }
  __syncthreads();

  // ---- phase 2: exponentiate in place, per-wave row sums
  for (int r = 0; r < 16; ++r) {
    float gm = wmx[r];
#pragma unroll
    for (int ww = 1; ww < 8; ++ww) gm = fmaxf(gm, wmx[ww * 16 + r]);
    float ps = 0.f;
    for (int cc = w * 512 + lane; cc < w * 512 + 512; cc += 32) {
      const float e2 = expf(P[r * LTOK + cc] - gm);
      P[r * LTOK + cc] = e2;
      ps += e2;
    }
    for (int off = 1; off < 32; off <<= 1) ps += __shfl_xor(ps, off, 32);
    if (lane == 0) wsm[w * 16 + r] = ps;
  }
  __syncthreads();

  // ---- phase 3: O = P @ V, each wave owns 4 e-tiles (64 columns)
  v8f acc[4];
#pragma unroll
  for (int t = 0; t < 4; ++t) acc[t] = (v8f){0.f,0.f,0.f,0.f,0.f,0.f,0.f,0.f};

  {
    const float* pp = P + (size_t)lid * LTOK + 2 * half;               // LDS
    const float* vp = Vf + ((size_t)(2 * half) * NB + n) * E + w * 64 + lid;
    const size_t vstep = (size_t)4 * NB * E;
#pragma unroll 2
    for (int sb = 0; sb < LTOK; sb += 4) {
      const v2f a = *(const v2f*)pp;
#pragma unroll
      for (int t = 0; t < 4; ++t) {
        v2f b;
        b.x = vp[t * 16];
        b.y = vp[(size_t)NB * E + t * 16];
        acc[t] = wmma4(a, b, acc[t]);
      }
      pp += 4;  vp += vstep;
    }
  }

  float gs[8];
#pragma unroll
  for (int v = 0; v < 8; ++v) {
    const int r = v + 8 * half;
    float s = 0.f;
#pragma unroll
    for (int ww = 0; ww < 8; ++ww) s += wsm[ww * 16 + r];
    gs[v] = 1.0f / s;
  }
#pragma unroll
  for (int t = 0; t < 4; ++t) {
    const int ec = w * 64 + t * 16 + lid;
#pragma unroll
    for (int v = 0; v < 8; ++v) {
      const int r = v + 8 * half;
      out[((size_t)(l0 + r) * NB + n) * E + ec] = acc[t][v] * gs[v];
    }
  }
}

// ---------------------------------------------------------------- host driver
extern "C" void kernel_launch(void* const* d_in, const int* in_sizes, int n_in,
                              void* d_out, int out_size, void* d_ws, size_t ws_size,
                              hipStream_t stream)
{
  (void)in_sizes; (void)n_in; (void)out_size; (void)ws_size;
  const float* queries = (const float*)d_in[0];
  const float* keys    = (const float*)d_in[1];
  const float* values  = (const float*)d_in[2];
  const float* slotv   = (const float*)d_in[3];
  const float* sWq = (const float*)d_in[4];
  const float* sWk = (const float*)d_in[5];
  const float* sWv = (const float*)d_in[6];
  const float* Wq  = (const float*)d_in[7];
  const float* Wk  = (const float*)d_in[8];
  const float* Wv  = (const float*)d_in[9];
  const float* l1w = (const float*)d_in[10];
  const float* l1b = (const float*)d_in[11];
  const float* l2w = (const float*)d_in[12];
  const float* l2b = (const float*)d_in[13];
  const float* n1g = (const float*)d_in[14];
  const float* n1b = (const float*)d_in[15];
  const float* n2g = (const float*)d_in[16];
  const float* n2b = (const float*)d_in[17];

  const size_t NLE = (size_t)LTOK * NB * E;      // 8,388,608
  float* outO    = (float*)d_out;                // out        [L,N,E]
  float* outQ    = outO + NLE;                   // Q          [L,N,E]
  float* outK    = outQ + NLE;                   // K          [L,N,E]
  float* outS    = outK + NLE;                   // slots      [12,N,E]
  float* outSA   = outS + (size_t)MS * NB * E;   // slot_assign[N,12,L]
  float* outSATT = outSA + (size_t)NB * MS * LTOK; // slot_att [N,L,12]

  float* ws = (float*)d_ws;
  float* Ks     = ws;                 // [L,N,E]  loop-invariant slot keys
  float* Vs     = Ks    + NLE;        // [L,N,E]  loop-invariant slot values
  float* Vf     = Vs    + NLE;        // [L,N,E]  final values
  float* slotsP = Vf    + NLE;        // [N,16,E] padded slots
  float* qs     = slotsP + (size_t)NB * MPAD * E;
  float* attb   = qs    + (size_t)NB * MPAD * E;   // [N,16,L]
  float* attsc  = attb  + (size_t)NB * MPAD * LTOK;
  float* srcA   = attsc + (size_t)NB * MPAD * LTOK;
  float* srcN   = srcA  + (size_t)NB * MPAD * E;
  float* ffnh   = srcN  + (size_t)NB * MPAD * E;   // [N,16,DFF]
  float* ffno   = ffnh  + (size_t)NB * MPAD * DFF;

  const dim3 w32(32);
  const dim3 gBig(E / 64, (LTOK * NB) / 16, 1);    // 16x64 blocks per wave

  // hoisted & final projections: X[16384,512] @ W[512,512].T  (5 GEMMs)
  wmma_gemm_wt<<<gBig, w32, 0, stream>>>(queries, sWk, Ks,   nullptr, nullptr, E, E, E, E, 1.0f,   0);
  wmma_gemm_wt<<<gBig, w32, 0, stream>>>(queries, sWv, Vs,   nullptr, nullptr, E, E, E, E, 1.0f,   0);
  wmma_gemm_wt<<<gBig, w32, 0, stream>>>(queries, Wq,  outQ, nullptr, nullptr, E, E, E, E, SCALING,0);
  wmma_gemm_wt<<<gBig, w32, 0, stream>>>(keys,    Wk,  outK, nullptr, nullptr, E, E, E, E, 1.0f,   0);
  wmma_gemm_wt<<<gBig, w32, 0, stream>>>(values,  Wv,  Vf,   nullptr, nullptr, E, E, E, E, 1.0f,   0);

  init_slots<<<(NB * MPAD * E + 255) / 256, 256, 0, stream>>>(slotv, slotsP);

  for (int cyc = 0; cyc < 3; ++cyc) {
    // Qs = slots @ sWq.T * SCALING   (64 padded rows)
    wmma_gemm_wt<<<dim3(E/64, (NB*MPAD)/16, 1), w32, 0, stream>>>(
        slotsP, sWq, qs, nullptr, nullptr, E, E, E, E, SCALING, 0);
    slot_scores <<<dim3(LTOK/256, NB), 256, 0, stream>>>(qs, Ks, attb);
    softmax_rows<<<NB * MS, 256, 0, stream>>>(attb, attsc);       // over tokens
    slot_assign_k<<<dim3(LTOK/256, NB), 256, 0, stream>>>(attb, outSA); // over slots
    // src = slots + att_scores @ V   (batched over n: B = Vs[n], K-major)
    wmma_gemm_nn_b<<<dim3(E/16, 1, NB), w32, 0, stream>>>(
        attsc, Vs, srcA, slotsP,
        LTOK, LTOK, NB*E, E,
        (long)MPAD*LTOK, (long)E, (long)MPAD*E, (long)MPAD*E);
    layernorm_k<<<NB * MPAD, 256, 0, stream>>>(srcA, n1g, n1b, srcN);
    // ffn1: gelu(srcN @ l1w.T + l1b)
    wmma_gemm_wt<<<dim3(DFF/64, (NB*MPAD)/16, 1), w32, 0, stream>>>(
        srcN, l1w, ffnh, l1b, nullptr, E, E, E, DFF, 1.0f, 1);
    // ffn2: h @ l2w.T + l2b + srcN (residual)
    wmma_gemm_wt<<<dim3(E/64, (NB*MPAD)/16, 1), w32, 0, stream>>>(
        ffnh, l2w, ffno, l2b, srcN, DFF, DFF, DFF, E, 1.0f, 0);
    layernorm_k<<<NB * MPAD, 256, 0, stream>>>(ffno, n2g, n2b, slotsP);
  }

  slot_att_k<<<dim3(LTOK/256, NB), 256, 0, stream>>>(outQ, slotsP, outSATT);

  const size_t shmem = (size_t)(MPAD*LTOK + MPAD*E + MPAD*MS + 128 + 128) * sizeof(float);
  hipFuncSetAttribute((const void*)flash_attn,
                      hipFuncAttributeMaxDynamicSharedMemorySize, (int)shmem);
  flash_attn<<<dim3(LTOK/16, NB), 256, shmem, stream>>>(outQ, outK, Vf, outSATT, outSA, outO);

  slots_out_k<<<(MS * NB * E + 255) / 256, 256, 0, stream>>>(slotsP, outS);
}